// ECE_2_52338471469097
// MI455X (gfx1250) — compile-verified
//
#include <hip/hip_runtime.h>
#include <hip/hip_bf16.h>

typedef float vf4 __attribute__((ext_vector_type(4)));
typedef int   vi4 __attribute__((ext_vector_type(4)));
typedef float v2f __attribute__((ext_vector_type(2)));
typedef float v8f __attribute__((ext_vector_type(8)));

#define NBINS 15
#define TPB   256
#define NB_MAX 2048
#define NSLOT 32        // 8 waves * 4 lane-replicas
#define PAD   16        // bins padded 15 -> 16 (bank-friendly)
#define WSCOL 48        // 45 partial columns padded to 48 (3 x 16 for WMMA)

__device__ __forceinline__ int bin_of(float c) {
    int b = (int)__builtin_ceilf(c * 15.0f) - 1;
    b = (b < 0) ? 0 : b;
    b = (b > NBINS - 1) ? (NBINS - 1) : b;
    return b;
}

__global__ __launch_bounds__(TPB)
void ece_partial(const float* __restrict__ conf, const int* __restrict__ pred,
                 const int* __restrict__ lab, float* __restrict__ ws, long long n) {
    __shared__ float        hconf[NSLOT * PAD];
    __shared__ unsigned int hca[NSLOT * PAD];   // cnt in [15:0], acc in [31:16]
    const int tid = threadIdx.x;
    for (int j = tid; j < NSLOT * PAD; j += TPB) { hconf[j] = 0.0f; hca[j] = 0u; }
    __syncthreads();

    const int slot = ((tid >> 5) << 2) | (tid & 3);   // wave*4 + lane%4 replica
    const int base = slot * PAD;

    const long long stride = (long long)gridDim.x * TPB * 4;
    long long i = ((long long)blockIdx.x * TPB + tid) * 4;
    for (; i + 3 < n; i += stride) {
        vf4 c = __builtin_nontemporal_load((const vf4*)(conf + i));
        vi4 p = __builtin_nontemporal_load((const vi4*)(pred + i));
        vi4 l = __builtin_nontemporal_load((const vi4*)(lab + i));
#pragma unroll
        for (int k = 0; k < 4; ++k) {
            const float cv = c[k];
            const int b = bin_of(cv);
            atomicAdd(&hconf[base + b], cv);
            const unsigned int inc = 1u | ((p[k] == l[k]) ? 0x10000u : 0u);
            atomicAdd(&hca[base + b], inc);
        }
    }
    // scalar tail (n % 4), handled by one thread
    if (blockIdx.x == 0 && tid == 0) {
        for (long long q = n & ~3LL; q < n; ++q) {
            const float cv = conf[q];
            const int b = bin_of(cv);
            atomicAdd(&hconf[b], cv);
            atomicAdd(&hca[b], 1u | ((pred[q] == lab[q]) ? 0x10000u : 0u));
        }
    }
    __syncthreads();

    // columns: [0..14]=conf_sum, [15..29]=acc_sum, [30..44]=count, [45..47]=0 pad
    if (tid < WSCOL) {
        float v = 0.0f;
        if (tid < 15) {
            float s = 0.0f;
            for (int s2 = 0; s2 < NSLOT; ++s2) s += hconf[s2 * PAD + tid];
            v = s;
        } else if (tid < 30) {
            unsigned int a = 0; const int b = tid - 15;
            for (int s2 = 0; s2 < NSLOT; ++s2) a += (hca[s2 * PAD + b] >> 16);
            v = (float)a;
        } else if (tid < 45) {
            unsigned int ct = 0; const int b = tid - 30;
            for (int s2 = 0; s2 < NSLOT; ++s2) ct += (hca[s2 * PAD + b] & 0xFFFFu);
            v = (float)ct;
        }
        ws[(long long)blockIdx.x * WSCOL + tid] = v;
    }
}

__global__ __launch_bounds__(TPB)
void ece_finalize(const float* __restrict__ ws, float* __restrict__ out,
                  int nb, long long n) {
    __shared__ float colsum[WSCOL];
    const int tid = threadIdx.x;
    if (tid < WSCOL) colsum[tid] = 0.0f;
    __syncthreads();

    const int w = tid >> 5, lane = tid & 31;
#if __has_builtin(__builtin_amdgcn_wmma_f32_16x16x4_f32)
    // Ones-matrix WMMA column-sum: D[m][n] = sum_k B[k][n]. 6 waves cover
    // 3 column groups x 2 halves of the block range; each WMMA reduces 4 blocks.
    if (w < 6) {
        const int g = w >> 1, half = w & 1;
        const int nbHalf = nb >> 1;
        const int blkBase = half * nbHalf;
        const int ncol = lane & 15, krow = lane >> 4;   // N=lane%16 (documented)
        v2f a; a[0] = 1.0f; a[1] = 1.0f;                // all-ones A (16x4)
        v8f acc = {};
        for (int b0 = 0; b0 < nbHalf; b0 += 4) {
            v2f bm;
            bm[0] = ws[(long long)(blkBase + b0 + krow) * WSCOL + g * 16 + ncol];
            bm[1] = ws[(long long)(blkBase + b0 + 2 + krow) * WSCOL + g * 16 + ncol];
            acc = __builtin_amdgcn_wmma_f32_16x16x4_f32(false, a, false, bm,
                                                        (short)0, acc, false, false);
        }
        if (lane < 16) atomicAdd(&colsum[g * 16 + ncol], acc[0]);  // row 0 of D
    }
#else
    if (tid < 45) {
        float s = 0.0f;
        for (int b = 0; b < nb; ++b) s += ws[(long long)b * WSCOL + tid];
        colsum[tid] = s;
    }
#endif
    __syncthreads();

    if (tid == 0) {
        const float nf = (float)n;
        float ece = 0.0f;
        for (int b = 0; b < NBINS; ++b) {
            const float cs  = colsum[b];
            const float as  = colsum[15 + b];
            const float cnt = colsum[30 + b];
            const float safe = fmaxf(cnt, 1.0f);
            const float per  = (cnt > 0.0f)
                             ? fabsf(cs / safe - as / safe) * (cnt / nf) : 0.0f;
            ece += per;
        }
        out[0] = ece;
    }
}

extern "C" void kernel_launch(void* const* d_in, const int* in_sizes, int n_in,
                              void* d_out, int out_size, void* d_ws, size_t ws_size,
                              hipStream_t stream) {
    const float* conf = (const float*)d_in[0];
    const int*   pred = (const int*)d_in[1];
    const int*   lab  = (const int*)d_in[2];
    float* out = (float*)d_out;
    float* ws  = (float*)d_ws;
    const long long n = (long long)in_sizes[0];

    int nb = NB_MAX;
    const long long maxnb = (long long)(ws_size / (WSCOL * sizeof(float)));
    if (maxnb < nb) nb = (int)(maxnb & ~7LL);   // keep nb a multiple of 8
    if (nb < 8) nb = 8;

    ece_partial <<<nb, TPB, 0, stream>>>(conf, pred, lab, ws, n);
    ece_finalize<<<1, TPB, 0, stream>>>(ws, out, nb, n);
}